// OneHopGraphAttention_2697239461974
// MI455X (gfx1250) — compile-verified
//
#include <hip/hip_runtime.h>
#include <math.h>

#define N_NODES 50000
#define N_EDGES 800000
#define D 64
#define H 2
#define NEG_SLOPE 0.2f

typedef __attribute__((ext_vector_type(2))) float v2f;
typedef __attribute__((ext_vector_type(8))) float v8f;

// Monotone order-preserving float<->uint mapping for atomicMax-based segment max.
__device__ __forceinline__ unsigned fenc(float f) {
    unsigned u = __float_as_uint(f);
    return u ^ ((unsigned)((int)u >> 31) | 0x80000000u);
}
__device__ __forceinline__ float fdec(unsigned k) {
    unsigned u = (k & 0x80000000u) ? (k ^ 0x80000000u) : ~k;
    return __uint_as_float(u);
}

// ---------------------------------------------------------------------------
// 0) Initialize workspace: agg = 0, segsum = 0, segmax = encoded(-inf) (key 0)
// ---------------------------------------------------------------------------
__global__ void init_ws(float* __restrict__ segsum, unsigned* __restrict__ segmax,
                        float* __restrict__ agg) {
    int i = blockIdx.x * blockDim.x + threadIdx.x;
    if (i < N_NODES * H * D) agg[i] = 0.0f;
    if (i < N_NODES * H) { segmax[i] = 0u; segsum[i] = 0.0f; }
}

// ---------------------------------------------------------------------------
// 1) Per-node attention projections (separable logits):
//    a_dst[n,h] = w_att[h,0:64]  . nodes[n]
//    a_src[n,h] = w_att[h,64:128]. nodes[n] + b_att[h]
//    One wave32 per node; butterfly reduction via shfl_xor.
// ---------------------------------------------------------------------------
__global__ void node_proj(const float* __restrict__ nodes, const float* __restrict__ w_att,
                          const float* __restrict__ b_att, float* __restrict__ a_dst,
                          float* __restrict__ a_src) {
    int wave = (int)((blockIdx.x * blockDim.x + threadIdx.x) >> 5);
    int lane = threadIdx.x & 31;
    if (wave >= N_NODES) return;
    const float* x = nodes + (size_t)wave * D;
    float x0 = x[lane];
    float x1 = x[lane + 32];
#pragma unroll
    for (int h = 0; h < H; ++h) {
        const float* w = w_att + h * (2 * D);
        float dpart = x0 * w[lane]      + x1 * w[lane + 32];
        float spart = x0 * w[64 + lane] + x1 * w[96 + lane];
#pragma unroll
        for (int m = 16; m >= 1; m >>= 1) {
            dpart += __shfl_xor(dpart, m, 32);
            spart += __shfl_xor(spart, m, 32);
        }
        if (lane == 0) {
            a_dst[wave * H + h] = dpart;
            a_src[wave * H + h] = spart + b_att[h];
        }
    }
}

// ---------------------------------------------------------------------------
// 2) Segment max of leaky-relu logits over r = senders[]  (uint-encoded atomicMax)
// ---------------------------------------------------------------------------
__global__ void edge_max(const int* __restrict__ senders, const int* __restrict__ receivers,
                         const float* __restrict__ a_dst, const float* __restrict__ a_src,
                         unsigned* __restrict__ segmax) {
    int e = blockIdx.x * blockDim.x + threadIdx.x;
    if (e >= N_EDGES) return;
    int s = receivers[e];   // reference swaps: s = receivers input
    int r = senders[e];     //                  r = senders input (segment id)
#pragma unroll
    for (int h = 0; h < H; ++h) {
        float raw = a_dst[s * H + h] + a_src[r * H + h];
        float v = raw > 0.0f ? raw : NEG_SLOPE * raw;
        atomicMax(&segmax[r * H + h], fenc(v));
    }
}

// ---------------------------------------------------------------------------
// 3) z = exp(v - segmax[r]); store z per edge; segsum[r] += z (native f32 atomic)
// ---------------------------------------------------------------------------
__global__ void edge_sum(const int* __restrict__ senders, const int* __restrict__ receivers,
                         const float* __restrict__ a_dst, const float* __restrict__ a_src,
                         const unsigned* __restrict__ segmax, float* __restrict__ segsum,
                         float* __restrict__ zbuf) {
    int e = blockIdx.x * blockDim.x + threadIdx.x;
    if (e >= N_EDGES) return;
    int s = receivers[e];
    int r = senders[e];
#pragma unroll
    for (int h = 0; h < H; ++h) {
        float raw = a_dst[s * H + h] + a_src[r * H + h];
        float v = raw > 0.0f ? raw : NEG_SLOPE * raw;
        float m = fdec(segmax[r * H + h]);
        float z = __expf(v - m);
        zbuf[(size_t)e * H + h] = z;
        atomicAdd(&segsum[r * H + h], z);
    }
}

// ---------------------------------------------------------------------------
// 4) Scatter: agg[r, h*64+d] += att[e,h] * nodes[s,d].  One wave32 per edge;
//    32 lanes x 4 atomics cover the 128 columns. Per-edge scalars are
//    wave-uniform -> fully coalesced atomic bursts into L2.
// ---------------------------------------------------------------------------
__global__ void edge_scatter(const int* __restrict__ senders, const int* __restrict__ receivers,
                             const float* __restrict__ nodes, const float* __restrict__ zbuf,
                             const float* __restrict__ segsum, float* __restrict__ agg) {
    int wave = (int)((blockIdx.x * blockDim.x + threadIdx.x) >> 5);
    int lane = threadIdx.x & 31;
    if (wave >= N_EDGES) return;
    int s = receivers[wave];
    int r = senders[wave];
    const float* x = nodes + (size_t)s * D;
    float x0 = x[lane];
    float x1 = x[lane + 32];
    float att0 = zbuf[(size_t)wave * H + 0] / segsum[r * H + 0];
    float att1 = zbuf[(size_t)wave * H + 1] / segsum[r * H + 1];
    float* a = agg + (size_t)r * (H * D);
    atomicAdd(a + lane,      att0 * x0);
    atomicAdd(a + 32 + lane, att0 * x1);
    atomicAdd(a + 64 + lane, att1 * x0);
    atomicAdd(a + 96 + lane, att1 * x1);
}

// ---------------------------------------------------------------------------
// 5) out = relu(agg[N,128] @ w_up[128,64] + b_up) via V_WMMA_F32_16X16X4_F32.
//    One wave per 16-node row block; 4 column tiles of 16; K=128 in 32 steps.
//    A layout (16x4 f32): lanes 0-15 row=lane K={k0,k0+1}; lanes 16-31 K={k0+2,k0+3}.
//    B layout (4x16 f32): mirrored over columns.
//    C/D layout (16x16 f32, 8 VGPRs): lane<16 rows v, lane>=16 rows v+8.
// ---------------------------------------------------------------------------
__global__ void update_gemm(const float* __restrict__ agg, const float* __restrict__ w_up,
                            const float* __restrict__ b_up, float* __restrict__ out) {
    int wave = (int)((blockIdx.x * blockDim.x + threadIdx.x) >> 5);
    int lane = threadIdx.x & 31;
    int n0 = wave * 16;
    if (n0 >= N_NODES) return;          // wave-uniform: EXEC stays all-ones for WMMA
    int half = lane >> 4;               // 0: K pair {k0,k0+1}, 1: {k0+2,k0+3}
    int l = lane & 15;

    v8f acc0 = {}, acc1 = {}, acc2 = {}, acc3 = {};
    const float* arow = agg + (size_t)(n0 + l) * (H * D);

#pragma unroll
    for (int k0 = 0; k0 < H * D; k0 += 4) {
        int kk = k0 + half * 2;
        v2f a = *(const v2f*)(arow + kk);              // 8B-aligned pair load
        const float* bp = w_up + (size_t)kk * D;       // rows kk, kk+1 of w_up
        v2f b0, b1, b2, b3;
        b0.x = bp[l];       b0.y = bp[D + l];
        b1.x = bp[16 + l];  b1.y = bp[D + 16 + l];
        b2.x = bp[32 + l];  b2.y = bp[D + 32 + l];
        b3.x = bp[48 + l];  b3.y = bp[D + 48 + l];
        acc0 = __builtin_amdgcn_wmma_f32_16x16x4_f32(false, a, false, b0, (short)0, acc0, false, false);
        acc1 = __builtin_amdgcn_wmma_f32_16x16x4_f32(false, a, false, b1, (short)0, acc1, false, false);
        acc2 = __builtin_amdgcn_wmma_f32_16x16x4_f32(false, a, false, b2, (short)0, acc2, false, false);
        acc3 = __builtin_amdgcn_wmma_f32_16x16x4_f32(false, a, false, b3, (short)0, acc3, false, false);
    }

    float bb0 = b_up[l], bb1 = b_up[16 + l], bb2 = b_up[32 + l], bb3 = b_up[48 + l];
#pragma unroll
    for (int v = 0; v < 8; ++v) {
        int row = v + half * 8;
        float* orow = out + (size_t)(n0 + row) * D;
        orow[l]      = fmaxf(acc0[v] + bb0, 0.0f);
        orow[16 + l] = fmaxf(acc1[v] + bb1, 0.0f);
        orow[32 + l] = fmaxf(acc2[v] + bb2, 0.0f);
        orow[48 + l] = fmaxf(acc3[v] + bb3, 0.0f);
    }
}

// ---------------------------------------------------------------------------
extern "C" void kernel_launch(void* const* d_in, const int* in_sizes, int n_in,
                              void* d_out, int out_size, void* d_ws, size_t ws_size,
                              hipStream_t stream) {
    (void)in_sizes; (void)n_in; (void)out_size; (void)ws_size;
    const float* nodes     = (const float*)d_in[0];
    const int*   senders   = (const int*)d_in[1];
    const int*   receivers = (const int*)d_in[2];
    const float* w_att     = (const float*)d_in[3];
    const float* b_att     = (const float*)d_in[4];
    const float* w_up      = (const float*)d_in[5];
    const float* b_up      = (const float*)d_in[6];
    float* out = (float*)d_out;

    // Workspace layout (floats): a_dst | a_src | segmax | segsum | zbuf | agg
    float*    ws     = (float*)d_ws;
    float*    a_dst  = ws;                                   // N*H
    float*    a_src  = a_dst + (size_t)N_NODES * H;          // N*H
    unsigned* segmax = (unsigned*)(a_src + (size_t)N_NODES * H); // N*H
    float*    segsum = (float*)(segmax + (size_t)N_NODES * H);   // N*H
    float*    zbuf   = segsum + (size_t)N_NODES * H;         // E*H
    float*    agg    = zbuf + (size_t)N_EDGES * H;           // N*H*D

    init_ws<<<(N_NODES * H * D + 255) / 256, 256, 0, stream>>>(segsum, segmax, agg);
    node_proj<<<(N_NODES * 32 + 255) / 256, 256, 0, stream>>>(nodes, w_att, b_att, a_dst, a_src);
    edge_max<<<(N_EDGES + 255) / 256, 256, 0, stream>>>(senders, receivers, a_dst, a_src, segmax);
    edge_sum<<<(N_EDGES + 255) / 256, 256, 0, stream>>>(senders, receivers, a_dst, a_src,
                                                        segmax, segsum, zbuf);
    edge_scatter<<<(N_EDGES * 32 + 255) / 256, 256, 0, stream>>>(senders, receivers, nodes,
                                                                 zbuf, segsum, agg);
    int gemm_waves = N_NODES / 16;  // 3125 tiles of 16 nodes (exact)
    update_gemm<<<(gemm_waves * 32 + 255) / 256, 256, 0, stream>>>(agg, w_up, b_up, out);
}